// HeteroClassifier_76424648065943
// MI455X (gfx1250) — compile-verified
//
#include <hip/hip_runtime.h>

// HeteroGraphConv (4 relations) x2 + mean-pool + linear head for gfx1250.
//
// Roofline: propagation = ~4 GB of 1KB-row gathers (L2-resident, dominates);
// GEMMs = 105 GFLOP (negligible on WMMA). So: CSR-gather (no scatter atomics)
// + f32 WMMA (16x16x4) GEMMs with double-buffered async(global->LDS) staging.

#define N_NODES   100000
#define N_EDGES   500000
#define N_REL     4
#define DIM       256
#define N_CLASSES 32
#define N_GRAPHS  16

typedef float v2f __attribute__((ext_vector_type(2)));
typedef float v8f __attribute__((ext_vector_type(8)));

// gfx1250 async global->LDS copy (ASYNCcnt-tracked); inline asm is portable
// across ROCm-7.2 / amdgpu-toolchain clang builtin arity differences.
__device__ __forceinline__ void async_copy_b128(void* lds_dst, const void* gsrc) {
  asm volatile("global_load_async_to_lds_b128 %0, %1, off"
               :
               : "v"((unsigned)(unsigned long long)lds_dst),
                 "v"((unsigned long long)gsrc)
               : "memory");
}
__device__ __forceinline__ void wait_async0() {
  asm volatile("s_wait_asynccnt 0x0" ::: "memory");
}

// ---------------------------------------------------------------- utilities
__global__ void k_zero_u32(unsigned* __restrict__ p, int n) {
  int i = blockIdx.x * blockDim.x + threadIdx.x;
  if (i < n) p[i] = 0u;
}

// per-relation in/out degree counts
__global__ void k_degrees(const int* __restrict__ src, const int* __restrict__ dst,
                          unsigned* __restrict__ cnt_out, unsigned* __restrict__ cnt_in) {
  int i = blockIdx.x * blockDim.x + threadIdx.x;
  if (i >= N_REL * N_EDGES) return;
  int r = i / N_EDGES;
  atomicAdd(&cnt_out[r * N_NODES + src[i]], 1u);
  atomicAdd(&cnt_in [r * N_NODES + dst[i]], 1u);
}

// rs = rsqrt(clamp(deg, 1))
__global__ void k_rsqrt(const unsigned* __restrict__ cnt_out, const unsigned* __restrict__ cnt_in,
                        float* __restrict__ rs_out, float* __restrict__ rs_in) {
  int i = blockIdx.x * blockDim.x + threadIdx.x;
  if (i >= N_REL * N_NODES) return;
  unsigned co = cnt_out[i]; if (co < 1u) co = 1u;
  unsigned ci = cnt_in[i];  if (ci < 1u) ci = 1u;
  rs_out[i] = rsqrtf((float)co);
  rs_in[i]  = rsqrtf((float)ci);
}

// exclusive prefix sum of cnt_in per relation (one block per relation)
__global__ __launch_bounds__(1024) void k_scan(const unsigned* __restrict__ cnt_in,
                                               unsigned* __restrict__ offs) {
  const int r = blockIdx.x;
  __shared__ unsigned tmp[1024];
  __shared__ unsigned carry_s;
  if (threadIdx.x == 0) carry_s = 0u;
  __syncthreads();
  for (int base = 0; base < N_NODES; base += 1024) {
    int n = base + (int)threadIdx.x;
    unsigned v = (n < N_NODES) ? cnt_in[r * N_NODES + n] : 0u;
    tmp[threadIdx.x] = v;
    __syncthreads();
    for (int off = 1; off < 1024; off <<= 1) {
      unsigned t = (threadIdx.x >= (unsigned)off) ? tmp[threadIdx.x - off] : 0u;
      __syncthreads();
      tmp[threadIdx.x] += t;
      __syncthreads();
    }
    unsigned incl  = tmp[threadIdx.x];
    unsigned carry = carry_s;
    if (n < N_NODES) offs[r * (N_NODES + 1) + n] = carry + incl - v;
    __syncthreads();
    if (threadIdx.x == 1023) carry_s = carry + incl;
    __syncthreads();
  }
  if (threadIdx.x == 0) offs[r * (N_NODES + 1) + N_NODES] = carry_s;
}

// scatter edges into CSR-by-dst slots
__global__ void k_csr_fill(const int* __restrict__ src, const int* __restrict__ dst,
                           const unsigned* __restrict__ offs, unsigned* __restrict__ cursor,
                           int* __restrict__ csr_src) {
  int i = blockIdx.x * blockDim.x + threadIdx.x;
  if (i >= N_REL * N_EDGES) return;
  int r = i / N_EDGES;
  int d = dst[i];
  unsigned pos = offs[r * (N_NODES + 1) + d] + atomicAdd(&cursor[r * N_NODES + d], 1u);
  csr_src[(size_t)r * N_EDGES + pos] = src[i];
}

// h[n][c] = sum_r bias_r[c]
__global__ void k_init_bias(float* __restrict__ h, const float* __restrict__ b) {
  long i = (long)blockIdx.x * blockDim.x + threadIdx.x;
  if (i >= (long)N_NODES * DIM) return;
  int c = (int)(i & (DIM - 1));
  h[i] = b[c] + b[DIM + c] + b[2 * DIM + c] + b[3 * DIM + c];
}

// one wave32 per dst node: agg[n] = rs_in[n] * sum_{e in CSR(n)} relu?(feat[src_e]) * rs_out[src_e]
__global__ __launch_bounds__(256) void k_aggregate(
    const float* __restrict__ feat, const float* __restrict__ rs_out_r,
    const float* __restrict__ rs_in_r, const unsigned* __restrict__ offs_r,
    const int* __restrict__ csr_src_r, float* __restrict__ agg, int apply_relu) {
  const int wave = threadIdx.x >> 5;
  const int lane = threadIdx.x & 31;
  const int n = blockIdx.x * 8 + wave;
  if (n >= N_NODES) return;
  const unsigned e0 = offs_r[n], e1 = offs_r[n + 1];
  float acc[8];
#pragma unroll
  for (int j = 0; j < 8; ++j) acc[j] = 0.f;
  for (unsigned e = e0; e < e1; ++e) {
    const int s = csr_src_r[e];
    const float sc = rs_out_r[s];
    const float4* row = (const float4*)(feat + (long)s * DIM);
    float4 v0 = row[lane];
    float4 v1 = row[lane + 32];
    if (apply_relu) {
      v0.x = fmaxf(v0.x, 0.f); v0.y = fmaxf(v0.y, 0.f);
      v0.z = fmaxf(v0.z, 0.f); v0.w = fmaxf(v0.w, 0.f);
      v1.x = fmaxf(v1.x, 0.f); v1.y = fmaxf(v1.y, 0.f);
      v1.z = fmaxf(v1.z, 0.f); v1.w = fmaxf(v1.w, 0.f);
    }
    acc[0] += v0.x * sc; acc[1] += v0.y * sc; acc[2] += v0.z * sc; acc[3] += v0.w * sc;
    acc[4] += v1.x * sc; acc[5] += v1.y * sc; acc[6] += v1.z * sc; acc[7] += v1.w * sc;
  }
  const float si = rs_in_r[n];
  float4* orow = (float4*)(agg + (long)n * DIM);
  orow[lane]      = make_float4(acc[0] * si, acc[1] * si, acc[2] * si, acc[3] * si);
  orow[lane + 32] = make_float4(acc[4] * si, acc[5] * si, acc[6] * si, acc[7] * si);
}

// H[64x128 block tile] += A[N,256] @ W[256,256], f32 WMMA 16x16x4.
// Double-buffered global->LDS async staging (ASYNCcnt); 8 waves/block;
// wave -> row-tile (w&3), col-tiles 4*(w>>2)+{0..3}.
#define BM 64
#define BN 128
#define BK 16
#define APAD 4   // A row stride 20 floats: 16B-aligned rows, conflict-free col reads
#define BPAD 8   // B row stride 136 floats: 16B-aligned rows, half-waves 16 banks apart
__global__ __launch_bounds__(256) void k_gemm_wmma(const float* __restrict__ A,
                                                   const float* __restrict__ W,
                                                   float* __restrict__ H) {
  __shared__ float As[2][BM][BK + APAD];
  __shared__ float Bs[2][BK][BN + BPAD];
  const int tid  = threadIdx.x;
  const int lane = tid & 31;
  const int wave = tid >> 5;
  const long m0 = (long)blockIdx.x * BM;
  const int  n0 = blockIdx.y * BN;
  const int rt = wave & 3;   // row tile within block
  const int cg = wave >> 2;  // col-tile group (4 tiles each)
  v8f acc[4] = {};
  const int a_row = tid >> 2, a_col = (tid & 3) * 4;   // 64 x 16 A stage, 16B per thread
  const int b_row = tid >> 4, b_col = (tid & 15) * 8;  // 16 x 128 B stage, 32B per thread
  const int khalf = (lane < 16) ? 0 : 2;               // ISA f32 A/B half-wave K split
  const int ml = rt * 16 + (lane & 15);
  const int nl = lane & 15;

  long gm = m0 + a_row;
  if (gm > N_NODES - 1) gm = N_NODES - 1;  // clamp: junk rows never stored (epilogue guard)
  const float* aptr = A + gm * DIM + a_col;
  const float* wptr = W + (size_t)b_row * DIM + n0 + b_col;

  // issue async loads of K-chunk k0 into buffer p
  auto issue = [&](int k0, int p) {
    async_copy_b128(&As[p][a_row][a_col], aptr + k0);
    async_copy_b128(&Bs[p][b_row][b_col], wptr + k0 * DIM);
    async_copy_b128(&Bs[p][b_row][b_col + 4], wptr + k0 * DIM + 4);
  };

  issue(0, 0);
  for (int ic = 0; ic < DIM / BK; ++ic) {
    const int p = ic & 1;
    wait_async0();       // this wave's chunk-ic loads landed
    __syncthreads();     // => every wave's loads landed; prior reads of buf p^1 done
    if (ic + 1 < DIM / BK) issue((ic + 1) * BK, p ^ 1);  // overlap with compute below
#pragma unroll
    for (int kk = 0; kk < BK; kk += 4) {
      v2f a;
      a.x = As[p][ml][kk + khalf];
      a.y = As[p][ml][kk + khalf + 1];
#pragma unroll
      for (int c = 0; c < 4; ++c) {
        const int nn = (cg * 4 + c) * 16 + nl;
        v2f b;
        b.x = Bs[p][kk + khalf][nn];
        b.y = Bs[p][kk + khalf + 1][nn];
        acc[c] = __builtin_amdgcn_wmma_f32_16x16x4_f32(
            false, a, false, b, (short)0, acc[c], false, false);
      }
    }
  }
  // C/D layout: VGPR i -> row (i + (lane<16?0:8)), col lane&15
#pragma unroll
  for (int c = 0; c < 4; ++c) {
    const int nn = n0 + (cg * 4 + c) * 16 + nl;
#pragma unroll
    for (int i = 0; i < 8; ++i) {
      long row = m0 + rt * 16 + i + ((lane < 16) ? 0 : 8);
      if (row < N_NODES) H[row * DIM + nn] += acc[c][i];
    }
  }
}

// per-graph sums via per-block LDS accumulators (graph count is tiny: 16)
__global__ __launch_bounds__(256) void k_pool(const float* __restrict__ h2,
                                              const int* __restrict__ gid,
                                              float* __restrict__ pooled,
                                              float* __restrict__ pcnt) {
  __shared__ float acc[N_GRAPHS][DIM];
  __shared__ float cacc[N_GRAPHS];
  const int tid = threadIdx.x;
#pragma unroll
  for (int g = 0; g < N_GRAPHS; ++g) acc[g][tid] = 0.f;
  if (tid < N_GRAPHS) cacc[tid] = 0.f;
  __syncthreads();
  const int per = (N_NODES + gridDim.x - 1) / gridDim.x;
  long nb = (long)blockIdx.x * per;
  long ne = nb + per; if (ne > N_NODES) ne = N_NODES;
  for (long n = nb; n < ne; ++n) {
    const int g = gid[n];
    acc[g][tid] += h2[n * DIM + tid];
    if (tid == 0) cacc[g] += 1.f;
  }
  __syncthreads();
#pragma unroll
  for (int g = 0; g < N_GRAPHS; ++g) atomicAdd(&pooled[g * DIM + tid], acc[g][tid]);
  if (tid < N_GRAPHS) atomicAdd(&pcnt[tid], cacc[tid]);
}

// out[g][c] = bc[c] + (pooled[g]/max(cnt,1)) . Wc[:,c]   (16x32 outputs, trivial)
__global__ void k_head(const float* __restrict__ pooled, const float* __restrict__ pcnt,
                       const float* __restrict__ Wc, const float* __restrict__ bc,
                       float* __restrict__ out) {
  const int t = threadIdx.x;
  if (t >= N_GRAPHS * N_CLASSES) return;
  const int g = t >> 5;       // /N_CLASSES
  const int c = t & 31;       // %N_CLASSES
  const float inv = 1.f / fmaxf(pcnt[g], 1.f);
  float s = bc[c];
  for (int k = 0; k < DIM; ++k) s += pooled[g * DIM + k] * inv * Wc[k * N_CLASSES + c];
  out[g * N_CLASSES + c] = s;
}

extern "C" void kernel_launch(void* const* d_in, const int* in_sizes, int n_in,
                              void* d_out, int out_size, void* d_ws, size_t ws_size,
                              hipStream_t stream) {
  (void)in_sizes; (void)n_in; (void)out_size; (void)ws_size;
  const float* x   = (const float*)d_in[0];
  const float* W1  = (const float*)d_in[1];
  const float* b1  = (const float*)d_in[2];
  const float* W2  = (const float*)d_in[3];
  const float* b2  = (const float*)d_in[4];
  const float* Wc  = (const float*)d_in[5];
  const float* bc  = (const float*)d_in[6];
  const int*   src = (const int*)d_in[7];
  const int*   dst = (const int*)d_in[8];
  const int*   gid = (const int*)d_in[9];
  float* out = (float*)d_out;

  char* w = (char*)d_ws;
  auto take = [&](size_t bytes) -> char* {
    char* p = w; w += (bytes + 255) & ~(size_t)255; return p;
  };
  float*    agg     = (float*)take((size_t)N_NODES * DIM * 4);
  float*    h1      = (float*)take((size_t)N_NODES * DIM * 4);
  float*    h2      = (float*)take((size_t)N_NODES * DIM * 4);
  float*    rs_out  = (float*)take((size_t)N_REL * N_NODES * 4);
  float*    rs_in   = (float*)take((size_t)N_REL * N_NODES * 4);
  unsigned* cnt_out = (unsigned*)take((size_t)N_REL * N_NODES * 4);
  unsigned* cnt_in  = (unsigned*)take((size_t)N_REL * N_NODES * 4);
  unsigned* cursor  = (unsigned*)take((size_t)N_REL * N_NODES * 4);
  unsigned* offs    = (unsigned*)take((size_t)N_REL * (N_NODES + 1) * 4);
  int*      csr     = (int*)take((size_t)N_REL * N_EDGES * 4);
  float*    pooled  = (float*)take((size_t)(N_GRAPHS * DIM + N_GRAPHS) * 4);
  float*    pcnt    = pooled + N_GRAPHS * DIM;

  const int T = 256;
  const int nRN = N_REL * N_NODES;
  const int nRE = N_REL * N_EDGES;

  // CSR + degree prep (reused by both layers)
  k_zero_u32<<<(nRN + T - 1) / T, T, 0, stream>>>(cnt_out, nRN);
  k_zero_u32<<<(nRN + T - 1) / T, T, 0, stream>>>(cnt_in, nRN);
  k_zero_u32<<<(nRN + T - 1) / T, T, 0, stream>>>(cursor, nRN);
  k_zero_u32<<<(N_GRAPHS * DIM + N_GRAPHS + T - 1) / T, T, 0, stream>>>(
      (unsigned*)pooled, N_GRAPHS * DIM + N_GRAPHS);
  k_degrees<<<(nRE + T - 1) / T, T, 0, stream>>>(src, dst, cnt_out, cnt_in);
  k_rsqrt<<<(nRN + T - 1) / T, T, 0, stream>>>(cnt_out, cnt_in, rs_out, rs_in);
  k_scan<<<N_REL, 1024, 0, stream>>>(cnt_in, offs);
  k_csr_fill<<<(nRE + T - 1) / T, T, 0, stream>>>(src, dst, offs, cursor, csr);

  const long nd = (long)N_NODES * DIM;
  const dim3 ggrid((N_NODES + BM - 1) / BM, DIM / BN);
  const dim3 agrid((N_NODES + 7) / 8);

  // layer 1: h1 = sum_r (agg_r @ W1_r) + sum_r b1_r   (relu deferred to layer-2 gather)
  k_init_bias<<<(nd + T - 1) / T, T, 0, stream>>>(h1, b1);
  for (int r = 0; r < N_REL; ++r) {
    k_aggregate<<<agrid, 256, 0, stream>>>(x, rs_out + (size_t)r * N_NODES,
                                           rs_in + (size_t)r * N_NODES,
                                           offs + (size_t)r * (N_NODES + 1),
                                           csr + (size_t)r * N_EDGES, agg, 0);
    k_gemm_wmma<<<ggrid, 256, 0, stream>>>(agg, W1 + (size_t)r * DIM * DIM, h1);
  }
  // layer 2: gather applies relu(h1)
  k_init_bias<<<(nd + T - 1) / T, T, 0, stream>>>(h2, b2);
  for (int r = 0; r < N_REL; ++r) {
    k_aggregate<<<agrid, 256, 0, stream>>>(h1, rs_out + (size_t)r * N_NODES,
                                           rs_in + (size_t)r * N_NODES,
                                           offs + (size_t)r * (N_NODES + 1),
                                           csr + (size_t)r * N_EDGES, agg, 1);
    k_gemm_wmma<<<ggrid, 256, 0, stream>>>(agg, W2 + (size_t)r * DIM * DIM, h2);
  }

  k_pool<<<400, 256, 0, stream>>>(h2, gid, pooled, pcnt);
  k_head<<<1, 512, 0, stream>>>(pooled, pcnt, Wc, bc, out);
}